// Complex_CA_41188736368824
// MI455X (gfx1250) — compile-verified
//
#include <hip/hip_runtime.h>
#include <hip/hip_bf16.h>

#define GRID   34
#define NPX    (GRID*GRID)            // 1156
#define PADW   36
#define PAD2W  38
#define TILE   16
#define NTILES ((NPX + TILE - 1)/TILE) // 73
#define COLS   (NTILES*TILE)           // 1168
#define BATCH  1024
#define PPT    5

typedef __attribute__((ext_vector_type(2))) float v2f;
typedef __attribute__((ext_vector_type(8))) float v8f;

__device__ __forceinline__ float wave_sum_f(float v) {
#pragma unroll
    for (int off = 16; off > 0; off >>= 1) v += __shfl_xor(v, off, 32);
    return v;
}
__device__ __forceinline__ int wave_sum_i(int v) {
#pragma unroll
    for (int off = 16; off > 0; off >>= 1) v += __shfl_xor(v, off, 32);
    return v;
}

__global__ __launch_bounds__(256)
void ca_kernel(const float* __restrict__ cellIn, const float* __restrict__ foodIn,
               const float* __restrict__ w1, const float* __restrict__ b1,
               const float* __restrict__ w2, const float* __restrict__ b2,
               const float* __restrict__ w3, const float* __restrict__ b3,
               const int* __restrict__ stepsPtr, float* __restrict__ out)
{
    __shared__ float sC[4*PADW*PADW];     // padded cell, zero halo
    __shared__ float sX0[PADW*PADW];      // padded x0 for post-life maxpool
    __shared__ float sSc[NPX];            // scent
    __shared__ float sAct[16*COLS];       // 12-ch acts / h1 / h2 / delta(rows0-3),
                                          // padded to 16 rows: branch-free D stores
    __shared__ float sBc[PAD2W*PAD2W];    // binarized pre-cell, halo-2 zeros
    __shared__ float sBx[PAD2W*PAD2W];    // binarized post-x,   halo-2 zeros
    __shared__ float sWp[3*256];          // 3 weight mats padded to 16x16
    __shared__ float sBias[48];           // 3 bias vecs padded to 16
    __shared__ float sWg[19];             // 1D gaussian taps
    __shared__ int   sRedI[16];           // double-buffered int reduce
    __shared__ float sRedF[16];

    const int tid   = threadIdx.x;
    const int b     = blockIdx.x;
    const int lane  = tid & 31;
    const int widx  = tid >> 5;
    const int half  = lane >> 4;          // WMMA lane-half (K pair select)
    const int nsel  = lane & 15;          // WMMA N (out-ch) / M (pixel) index
    const int steps = stepsPtr[0];

    // ---- zero-init LDS (determinism + zero halos) ----
    for (int i = tid; i < 4*PADW*PADW; i += 256) sC[i]  = 0.f;
    for (int i = tid; i < PADW*PADW;   i += 256) sX0[i] = 0.f;
    for (int i = tid; i < 16*COLS;     i += 256) sAct[i] = 0.f;
    for (int i = tid; i < PAD2W*PAD2W; i += 256) { sBc[i] = 0.f; sBx[i] = 0.f; }
    for (int i = tid; i < 3*256;       i += 256) sWp[i] = 0.f;
    if (tid < 48) sBias[tid] = 0.f;
    __syncthreads();

    // ---- load weights (padded 12->16), biases, gaussian taps ----
    if (tid < 144) {
        sWp[0*256 + (tid/12)*16 + (tid%12)] = w1[tid];
        sWp[1*256 + (tid/12)*16 + (tid%12)] = w2[tid];
    }
    if (tid < 48) sWp[2*256 + (tid/12)*16 + (tid%12)] = w3[tid];
    if (tid < 12) { sBias[tid] = b1[tid]; sBias[16+tid] = b2[tid]; }
    if (tid < 4)  sBias[32+tid] = b3[tid];
    if (tid < 19) {
        float d = (float)tid * (19.0f/18.0f) - 9.5f;     // linspace(0,19,19)-9.5
        sWg[tid] = expf(-(d*d) / (2.0f * 4.75f * 4.75f));
    }
    // cell -> padded LDS
    const float* cb = cellIn + (size_t)b * 4 * NPX;
    for (int i = tid; i < 4*NPX; i += 256) {
        int ch = i / NPX, p = i - ch*NPX;
        int y = p / GRID, x = p - y*GRID;
        sC[ch*PADW*PADW + (y+1)*PADW + (x+1)] = cb[i];
    }
    // food -> scratch (reuse sAct row 0; row 1 holds the horizontal pass)
    const float* fb = foodIn + (size_t)b * NPX;
    for (int i = tid; i < NPX; i += 256) sAct[i] = fb[i];
    __syncthreads();

    // ---- scent = separable 19x19 gaussian conv of food (once) ----
    {
        float* fd = sAct; float* hb = sAct + COLS;
        for (int i = tid; i < NPX; i += 256) {
            int y = i / GRID, x = i - y*GRID;
            float s = 0.f;
#pragma unroll
            for (int j = 0; j < 19; ++j) {
                int xx = x + j - 9;
                if (xx >= 0 && xx < GRID) s += fd[y*GRID + xx] * sWg[j];
            }
            hb[i] = s;
        }
        __syncthreads();
        const float SCALE = 80.0f / (2.0f * 3.14159265358979323846f * 4.75f * 4.75f);
        for (int i = tid; i < NPX; i += 256) {
            int y = i / GRID, x = i - y*GRID;
            float s = 0.f;
#pragma unroll
            for (int j = 0; j < 19; ++j) {
                int yy = y + j - 9;
                if (yy >= 0 && yy < GRID) s += hb[yy*GRID + x] * sWg[j];
            }
            sSc[i] = s * SCALE;
        }
        __syncthreads();
    }

    // ---- preload WMMA B matrices (W^T chunks) and biases into registers ----
    // B 4x16 layout: VGPR0: lanes0-15 K=kb, lanes16-31 K=kb+2 ; VGPR1: K=kb+1 / kb+3
    v2f  Breg[3][3];
    float biasr[3];
#pragma unroll
    for (int l = 0; l < 3; ++l) {
        biasr[l] = sBias[l*16 + nsel];
#pragma unroll
        for (int c = 0; c < 3; ++c) {
            Breg[l][c].x = sWp[l*256 + nsel*16 + 4*c + 2*half];
            Breg[l][c].y = sWp[l*256 + nsel*16 + 4*c + 2*half + 1];
        }
    }

    float x0r[PPT], x1r[PPT], x2r[PPT], x3r[PPT];

    // ================= main 64-step loop, fully LDS-resident =================
    for (int s = 0; s < steps; ++s) {
        // channel 3 <- scent
        for (int k = 0; k < PPT; ++k) { int p = tid + k*256; if (p < NPX) {
            int y = p / GRID, x = p - y*GRID;
            sC[3*PADW*PADW + (y+1)*PADW + (x+1)] = sSc[p];
        }}
        __syncthreads();

        // phase 2: pre-life, binarize c0, sobel pair -> 12-ch activations
        unsigned preBits = 0;
        int livLocal = 0;
        for (int k = 0; k < PPT; ++k) { int p = tid + k*256; if (p < NPX) {
            int y = p / GRID, x = p - y*GRID;
            int pi = (y+1)*PADW + (x+1);
#pragma unroll
            for (int ch = 0; ch < 4; ++ch) {
                const float* cp = sC + ch*PADW*PADW;
                float n00=cp[pi-PADW-1], n01=cp[pi-PADW], n02=cp[pi-PADW+1];
                float n10=cp[pi-1],      n11=cp[pi],      n12=cp[pi+1];
                float n20=cp[pi+PADW-1], n21=cp[pi+PADW], n22=cp[pi+PADW+1];
                if (ch == 0) {
                    float mx = fmaxf(fmaxf(fmaxf(n00,n01),fmaxf(n02,n10)),
                                     fmaxf(fmaxf(n11,n12),fmaxf(fmaxf(n20,n21),n22)));
                    if (mx > 0.1f) preBits |= (1u << k);
                    bool alive = (n11 > 0.1f);
                    sBc[(y+2)*PAD2W + (x+2)] = alive ? 1.f : 0.f;
                    livLocal += alive ? 1 : 0;
                }
                float g1 = ((n02 - n00) + 2.f*(n12 - n10) + (n22 - n20)) * 0.125f;
                float g2 = ((n20 - n00) + 2.f*(n21 - n01) + (n22 - n02)) * 0.125f;
                sAct[ch*COLS + p]     = n11;
                sAct[(4+ch)*COLS + p] = g1;
                sAct[(8+ch)*COLS + p] = g2;
            }
        }}
        __syncthreads();

        // phase WMMA: 12->12->12->4 MLP in f32 on the matrix pipes.
        // D[16px x 16oc] += A[16px x 4k] * B[4k x 16oc], K=12 as 3 chunks.
        // Branch-free: all 16 D rows stored (rows 12-15 dead); layer-3 delta
        // lands in rows 0-3 of sAct after its A-loads (in-order LDS per wave).
        for (int t = widx; t < NTILES; t += 8) {     // wave-uniform; EXEC all 1s
            const int base = t * TILE;
            const int acol = base + nsel;            // A row M = lane&15
            const int drow = nsel*COLS + base + 8*half;
            v8f acc;
            // ---- layer 1 ----
#pragma unroll
            for (int v = 0; v < 8; ++v) acc[v] = biasr[0];
#pragma unroll
            for (int c = 0; c < 3; ++c) {
                v2f a;
                a.x = sAct[(4*c + 2*half)*COLS + acol];
                a.y = sAct[(4*c + 2*half + 1)*COLS + acol];
                acc = __builtin_amdgcn_wmma_f32_16x16x4_f32(
                          false, a, false, Breg[0][c], (short)0, acc, false, false);
            }
#pragma unroll
            for (int v = 0; v < 8; ++v) acc[v] = fmaxf(acc[v], 0.f);
#pragma unroll
            for (int v = 0; v < 8; ++v) sAct[drow + v] = acc[v];
            // ---- layer 2 ----
#pragma unroll
            for (int v = 0; v < 8; ++v) acc[v] = biasr[1];
#pragma unroll
            for (int c = 0; c < 3; ++c) {
                v2f a;
                a.x = sAct[(4*c + 2*half)*COLS + acol];
                a.y = sAct[(4*c + 2*half + 1)*COLS + acol];
                acc = __builtin_amdgcn_wmma_f32_16x16x4_f32(
                          false, a, false, Breg[1][c], (short)0, acc, false, false);
            }
#pragma unroll
            for (int v = 0; v < 8; ++v) acc[v] = fmaxf(acc[v], 0.f);
#pragma unroll
            for (int v = 0; v < 8; ++v) sAct[drow + v] = acc[v];
            // ---- layer 3 (no relu), delta -> rows 0..3 (others dead) ----
#pragma unroll
            for (int v = 0; v < 8; ++v) acc[v] = biasr[2];
#pragma unroll
            for (int c = 0; c < 3; ++c) {
                v2f a;
                a.x = sAct[(4*c + 2*half)*COLS + acol];
                a.y = sAct[(4*c + 2*half + 1)*COLS + acol];
                acc = __builtin_amdgcn_wmma_f32_16x16x4_f32(
                          false, a, false, Breg[2][c], (short)0, acc, false, false);
            }
#pragma unroll
            for (int v = 0; v < 8; ++v) sAct[drow + v] = acc[v];
        }
        __syncthreads();

        // phase 3a: x = c + delta; publish x0 for post-life maxpool
        for (int k = 0; k < PPT; ++k) { int p = tid + k*256; if (p < NPX) {
            int y = p / GRID, x = p - y*GRID;
            int pi = (y+1)*PADW + (x+1);
            float s0 = sC[pi]                 + sAct[p];
            x1r[k]   = sC[1*PADW*PADW + pi]   + sAct[1*COLS + p];
            x2r[k]   = sC[2*PADW*PADW + pi]   + sAct[2*COLS + p];
            x3r[k]   = sC[3*PADW*PADW + pi]   + sAct[3*COLS + p];
            sX0[pi] = s0;
        }}
        __syncthreads();

        // phase 3b: post-life, life mask, clips
        for (int k = 0; k < PPT; ++k) { int p = tid + k*256; if (p < NPX) {
            int y = p / GRID, x = p - y*GRID;
            int pi = (y+1)*PADW + (x+1);
            float m00=sX0[pi-PADW-1], m01=sX0[pi-PADW], m02=sX0[pi-PADW+1];
            float m10=sX0[pi-1],      m11=sX0[pi],      m12=sX0[pi+1];
            float m20=sX0[pi+PADW-1], m21=sX0[pi+PADW], m22=sX0[pi+PADW+1];
            float mx = fmaxf(fmaxf(fmaxf(m00,m01),fmaxf(m02,m10)),
                             fmaxf(fmaxf(m11,m12),fmaxf(fmaxf(m20,m21),m22)));
            bool life = (mx > 0.1f) && ((preBits >> k) & 1u);
            float mf = life ? 1.f : 0.f;
            x0r[k] = fminf(fmaxf(m11 * mf,  0.f),  1.f);   // clip(-10,10) then (0,1)
            x1r[k] = fminf(fmaxf(x1r[k]*mf, -10.f), 10.f);
            x2r[k] = fminf(fmaxf(x2r[k]*mf, -10.f), 10.f);
            x3r[k] = fminf(fmaxf(x3r[k]*mf, -10.f), 10.f);
        }}

        // phase 3c: exact k-th largest via bit-pattern binary search
        int rb = 0;
        {
            int lv = wave_sum_i(livLocal);
            if (lane == 0) sRedI[rb*8 + widx] = lv;
        }
        __syncthreads();
        int L = 0;
        for (int w = 0; w < 8; ++w) L += sRedI[rb*8 + w];
        rb ^= 1;
        int kidx = NPX - 1 - L; if (kidx < 0) kidx += NPX;   // JAX negative wrap
        const int Leff = NPX - 1 - kidx;
        unsigned lo = 0u, hi = 0x3F800000u;                  // values in [0,1]
        while (lo < hi) {                                    // block-uniform
            unsigned mid = lo + ((hi - lo) >> 1);
            float fm = __uint_as_float(mid);
            int cnt = 0;
            for (int k = 0; k < PPT; ++k) { int p = tid + k*256;
                if (p < NPX && x0r[k] > fm) ++cnt; }
            cnt = wave_sum_i(cnt);
            if (lane == 0) sRedI[rb*8 + widx] = cnt;
            __syncthreads();
            int tot = 0;
            for (int w = 0; w < 8; ++w) tot += sRedI[rb*8 + w];
            if (tot <= Leff) hi = mid; else lo = mid + 1;
            rb ^= 1;
        }
        const float thr = __uint_as_float(lo);               // == srt[kidx]

        // phase 3d: binarize thresholded x0
        for (int k = 0; k < PPT; ++k) { int p = tid + k*256; if (p < NPX) {
            int y = p / GRID, x = p - y*GRID;
            float v = (x0r[k] > thr) ? x0r[k] : 0.f;
            sBx[(y+2)*PAD2W + (x+2)] = (v > 0.1f) ? 1.f : 0.f;
        }}
        __syncthreads();

        // phase 3e: rulebreak sum-pools (3x3 & 5x5, zero-pad), commit new cell
        for (int k = 0; k < PPT; ++k) { int p = tid + k*256; if (p < NPX) {
            int y = p / GRID, x = p - y*GRID;
            int cb2 = (y+2)*PAD2W + (x+2);
            float cs3=0.f, cs5=0.f, xs3=0.f, xs5=0.f;
#pragma unroll
            for (int dy = -2; dy <= 2; ++dy)
#pragma unroll
                for (int dx = -2; dx <= 2; ++dx) {
                    float bcv = sBc[cb2 + dy*PAD2W + dx];
                    float bxv = sBx[cb2 + dy*PAD2W + dx];
                    cs5 += bcv; xs5 += bxv;
                    if (dy >= -1 && dy <= 1 && dx >= -1 && dx <= 1) { cs3 += bcv; xs3 += bxv; }
                }
            bool rule = (xs5 >= cs3) && (xs3 <= cs5);
            float x0t = (x0r[k] > thr) ? x0r[k] : 0.f;
            int pi = (y+1)*PADW + (x+1);
            sC[pi]               = rule ? x0t : 0.f;
            sC[1*PADW*PADW + pi] = x1r[k];
            sC[2*PADW*PADW + pi] = x2r[k];
            sC[3*PADW*PADW + pi] = x3r[k];
        }}
        __syncthreads();
    }

    // ================= outputs =================
    float* outCell = out;
    float* outFood = out + (size_t)BATCH * 4 * NPX;
    float* outTPV  = outFood + (size_t)BATCH * NPX;
    float* outLiv  = outTPV + BATCH;
    float* outTox  = outLiv + BATCH;

    float* ocb = outCell + (size_t)b * 4 * NPX;
    for (int i = tid; i < 4*NPX; i += 256) {
        int ch = i / NPX, p = i - ch*NPX;
        int y = p / GRID, x = p - y*GRID;
        ocb[i] = sC[ch*PADW*PADW + (y+1)*PADW + (x+1)];
    }
    float* ofb = outFood + (size_t)b * NPX;
    for (int i = tid; i < NPX; i += 256) ofb[i] = fb[i];

    float tp = 0.f; int lc = 0;
    for (int k = 0; k < PPT; ++k) { int p = tid + k*256; if (p < NPX) {
        int y = p / GRID, x = p - y*GRID;
        float v = sC[(y+1)*PADW + (x+1)];
        tp += v; lc += (v > 0.1f) ? 1 : 0;
    }}
    tp = wave_sum_f(tp); lc = wave_sum_i(lc);
    if (lane == 0) { sRedF[widx] = tp; sRedI[widx] = lc; }
    __syncthreads();
    if (tid == 0) {
        float T = 0.f; int C = 0;
        for (int w = 0; w < 8; ++w) { T += sRedF[w]; C += sRedI[w]; }
        outTPV[b] = T;
        outLiv[b] = (float)C;
        if (b == 0) outTox[0] = 0.f;
    }
}

extern "C" void kernel_launch(void* const* d_in, const int* in_sizes, int n_in,
                              void* d_out, int out_size, void* d_ws, size_t ws_size,
                              hipStream_t stream) {
    (void)in_sizes; (void)n_in; (void)out_size; (void)d_ws; (void)ws_size;
    ca_kernel<<<BATCH, 256, 0, stream>>>(
        (const float*)d_in[0],  // cell
        (const float*)d_in[1],  // food
        (const float*)d_in[2],  // w1
        (const float*)d_in[3],  // b1
        (const float*)d_in[4],  // w2
        (const float*)d_in[5],  // b2
        (const float*)d_in[6],  // w3
        (const float*)d_in[7],  // b3
        (const int*)d_in[8],    // steps
        (float*)d_out);
}